// AsymmetricLossCustomPrioritySmallTest_57243324121485
// MI455X (gfx1250) — compile-verified
//
#include <hip/hip_runtime.h>
#include <stdint.h>

#define B_ROWS   4096
#define C_COLS   9605
#define TILE     1024
#define NTHREADS 512
#define NT       ((C_COLS + TILE - 1) / TILE)   // 10 tiles per row
#define MASKW    ((C_COLS + 31) / 32)           // 301 dwords
#define EPSV     1e-8f
#define CLIPV    0.05f

typedef unsigned int v4u __attribute__((ext_vector_type(4)));
typedef int          v8i __attribute__((ext_vector_type(8)));
typedef int          v4i __attribute__((ext_vector_type(4)));

// ---------- elementwise loss (mult == 1) ----------
// y==1: log(max(s,eps)) * (1-s)            [gamma=1]
// y==0: log(max(min(1.05-s,1),eps)) * max(s-0.05,0)^4   [gamma=4, integer power]
__device__ __forceinline__ float loss_elem(float xv, float yv) {
    float s = __builtin_amdgcn_rcpf(1.0f + __expf(-xv));   // sigmoid
    bool pos = (yv != 0.0f);
    float xneg = fminf(1.0f + CLIPV - s, 1.0f);
    float larg = pos ? fmaxf(s, EPSV) : fmaxf(xneg, EPSV);
    float base = __logf(larg);
    float t  = fmaxf(s - CLIPV, 0.0f);
    float t2 = t * t;
    float w  = pos ? (1.0f - s) : (t2 * t2);
    return base * w;
}

// ---------- top-10 register list (sorted desc, tie -> smaller index) ----------
__device__ __forceinline__ bool better(float av, int ai, float bv, int bi) {
    return (av > bv) || (av == bv && ai < bi);
}

__device__ __forceinline__ void topk_insert(float (&tv)[10], int (&ti)[10], float v, int idx) {
    if (!better(v, idx, tv[9], ti[9])) return;
    tv[9] = v; ti[9] = idx;
#pragma unroll
    for (int k = 9; k > 0; --k) {
        if (better(tv[k], ti[k], tv[k - 1], ti[k - 1])) {
            float fv = tv[k]; tv[k] = tv[k - 1]; tv[k - 1] = fv;
            int   ii = ti[k]; ti[k] = ti[k - 1]; ti[k - 1] = ii;
        }
    }
}

// ---------- TDM issue: one row-segment (2D descriptor, 1 x TILE tile) ----------
// tensor_dim0 = remaining elements in the row, so the tail tile zero-fills LDS.
__device__ __forceinline__ void tdm_issue(const float* gbase, unsigned ldsOff, int col0) {
    uint64_t ga  = (uint64_t)(uintptr_t)(gbase + col0);
    unsigned rem = (unsigned)(C_COLS - col0);
    v4u g0;
    g0[0] = 1u;                                   // count=1 (valid user descriptor)
    g0[1] = ldsOff;                               // lds_addr (bytes)
    g0[2] = (unsigned)ga;                         // global_addr[31:0]
    g0[3] = (unsigned)(ga >> 32) | (2u << 30);    // global_addr[56:32] | type=2
    v8i g1;
    g1[0] = (int)(2u << 16);                      // wg_mask=0, data_size=2 (4B)
    g1[1] = (int)((rem & 0xFFFFu) << 16);         // tensor_dim0[15:0]
    g1[2] = (int)((rem >> 16) | (1u << 16));      // tensor_dim0[31:16], tensor_dim1=1
    g1[3] = (int)((unsigned)TILE << 16);          // tile_dim0 = TILE
    g1[4] = 1;                                    // tile_dim1 = 1, tile_dim2 = 0
    g1[5] = (int)rem;                             // tensor_dim0_stride (unused: 1 row)
    g1[6] = 0;
    g1[7] = 0;
    v4i z4 = {0, 0, 0, 0};
    v8i z8 = {0, 0, 0, 0, 0, 0, 0, 0};
    __builtin_amdgcn_tensor_load_to_lds(g0, g1, z4, z4, z8, 0);
}

__global__ void __launch_bounds__(NTHREADS)
asl_main(const float* __restrict__ x, const float* __restrict__ y,
         const unsigned char* __restrict__ wlmask,
         const int* __restrict__ ci, const int* __restrict__ ri,
         const int* __restrict__ di, float* __restrict__ out) {
    __shared__ float    sx[2][TILE];
    __shared__ float    sy[2][TILE];
    __shared__ unsigned smask[MASKW];
    __shared__ float    mtv[NTHREADS / 2][10];
    __shared__ int      mti[NTHREADS / 2][10];
    __shared__ float    s_wl, s_sum;

    const int    tid    = threadIdx.x;
    const int    r      = blockIdx.x;
    const size_t rowoff = (size_t)r * C_COLS;

    // LDS byte offsets of the staging buffers (flat LDS addr -> low 32 bits = offset)
    const unsigned offX0 = (unsigned)(uintptr_t)&sx[0][0];
    const unsigned offX1 = (unsigned)(uintptr_t)&sx[1][0];
    const unsigned offY0 = (unsigned)(uintptr_t)&sy[0][0];
    const unsigned offY1 = (unsigned)(uintptr_t)&sy[1][0];

    // Kick off TDM double-buffer prologue from wave 0 only (EXEC ignored by TDM;
    // other waves must not reach the instruction).
    if (tid < 32) {
        tdm_issue(x + rowoff, offX0, 0);
        tdm_issue(y + rowoff, offY0, 0);
        tdm_issue(x + rowoff, offX1, TILE);
        tdm_issue(y + rowoff, offY1, TILE);
    }

    // Build whitelist-index bitmask + init accumulators (overlaps with TDM fill)
    for (int w = tid; w < MASKW; w += NTHREADS) smask[w] = 0u;
    if (tid == 0) { s_wl = 0.0f; s_sum = 0.0f; }
    __syncthreads();
    if (tid < 210) {
        int idx = (tid < 70) ? ci[tid] : (tid < 140 ? ri[tid - 70] : di[tid - 140]);
        atomicOr(&smask[idx >> 5], 1u << (idx & 31));
    }

    float tv[10]; int ti[10];
#pragma unroll
    for (int k = 0; k < 10; ++k) { tv[k] = -__builtin_inff(); ti[k] = 0x7FFFFFFF; }
    float lsum = 0.0f, wlh = 0.0f;

    for (int t = 0; t < NT; ++t) {
        const int b = t & 1;
        if (tid < 32) {   // wave 0 owns TENSORcnt; in-order -> <=2 means tile t done
            if (t + 1 < NT) __builtin_amdgcn_s_wait_tensorcnt(2);
            else            __builtin_amdgcn_s_wait_tensorcnt(0);
        }
        __syncthreads();                       // tile t visible to all waves

        const int col0 = t * TILE;
#pragma unroll
        for (int i = tid; i < TILE; i += NTHREADS) {
            int c = col0 + i;
            if (c < C_COLS) {
                float xv = sx[b][i];
                float yv = sy[b][i];
                lsum += loss_elem(xv, yv);
                if ((smask[c >> 5] >> (c & 31)) & 1u) wlh += yv;
                topk_insert(tv, ti, xv, c);
            }
        }
        __syncthreads();                       // everyone done reading buffer b
        if (tid < 32 && t + 2 < NT) {          // refill buffer b with tile t+2
            tdm_issue(x + rowoff, b ? offX1 : offX0, (t + 2) * TILE);
            tdm_issue(y + rowoff, b ? offY1 : offY0, (t + 2) * TILE);
        }
    }

    // row-level scalar reductions (LDS float atomics: ds_add_f32)
    atomicAdd(&s_wl, wlh);
    atomicAdd(&s_sum, lsum);

    // tournament merge of per-thread top-10 lists: 512 -> 1
    for (int off = NTHREADS / 2; off >= 1; off >>= 1) {
        if (tid >= off && tid < 2 * off) {
#pragma unroll
            for (int k = 0; k < 10; ++k) { mtv[tid - off][k] = tv[k]; mti[tid - off][k] = ti[k]; }
        }
        __syncthreads();
        if (tid < off) {
#pragma unroll
            for (int k = 0; k < 10; ++k) topk_insert(tv, ti, mtv[tid][k], mti[tid][k]);
        }
        __syncthreads();
    }

    if (tid == 0) {
        const bool gt_none = (s_wl == 0.0f);
        float extra = 0.0f;
#pragma unroll
        for (int k = 0; k < 10; ++k) {
            int   c  = ti[k];
            float yv = y[rowoff + c];
            bool  wl = (wlmask[c] != 0);
            bool  pen = wl ? (yv == 0.0f) : gt_none;
            if (pen) extra += loss_elem(tv[k], yv);   // (ALPHA3-1) * elem == elem
        }
        atomicAdd(out, -(s_sum + extra));
    }
}

__global__ void zero_out(float* out) { out[0] = 0.0f; }

extern "C" void kernel_launch(void* const* d_in, const int* in_sizes, int n_in,
                              void* d_out, int out_size, void* d_ws, size_t ws_size,
                              hipStream_t stream) {
    (void)in_sizes; (void)n_in; (void)out_size; (void)d_ws; (void)ws_size;
    const float*         x  = (const float*)d_in[0];
    const float*         y  = (const float*)d_in[1];
    const unsigned char* wl = (const unsigned char*)d_in[2];
    const int*           ci = (const int*)d_in[3];
    const int*           ri = (const int*)d_in[4];
    const int*           di = (const int*)d_in[5];
    float* out = (float*)d_out;

    zero_out<<<1, 1, 0, stream>>>(out);
    asl_main<<<B_ROWS, NTHREADS, 0, stream>>>(x, y, wl, ci, ri, di, out);
}